// UnsupervisedTriphard_23794118820055
// MI455X (gfx1250) — compile-verified
//
#include <hip/hip_runtime.h>
#include <hip/hip_bf16.h>

typedef __attribute__((ext_vector_type(2))) float v2f;
typedef __attribute__((ext_vector_type(8))) float v8f;

#define NPTS 8192
#define DIM  128
#define NTILE 512          // 8192 / 16 column tiles
#define WAVES 8            // waves per block

// ---------------------------------------------------------------------------
// Sorted ascending top-5 (value, index), lexicographic tie-break (stable top_k)
// ---------------------------------------------------------------------------
__device__ __forceinline__ void top5_insert(float (&tv)[5], int (&ti)[5],
                                            float d, int idx) {
    if (d > tv[4] || (d == tv[4] && idx >= ti[4])) return;
    float v = d; int id = idx;
#pragma unroll
    for (int s = 0; s < 5; ++s) {
        bool better = (v < tv[s]) || (v == tv[s] && id < ti[s]);
        float ov = tv[s]; int oi = ti[s];
        if (better) { tv[s] = v; ti[s] = id; v = ov; id = oi; }
    }
}

// ---------------------------------------------------------------------------
// Kernel 1: squared row norms. One wave32 per row (lane l -> float4 at 4*l).
// ---------------------------------------------------------------------------
__global__ __launch_bounds__(256) void triphard_sqnorm(
    const float* __restrict__ inp, float* __restrict__ sq) {
    const int w = threadIdx.x >> 5, lane = threadIdx.x & 31;
    const int row = blockIdx.x * 8 + w;
    const float4 v = *(const float4*)(inp + row * DIM + lane * 4);
    float s = v.x * v.x + v.y * v.y + v.z * v.z + v.w * v.w;
#pragma unroll
    for (int m = 16; m >= 1; m >>= 1) s += __shfl_xor(s, m, 32);
    if (lane == 0) sq[row] = s;
}

// ---------------------------------------------------------------------------
// Kernel 2: fused Gram + top-5 + triplet hinge per 16-row tile.
//   block = 256 (8 waves), grid = 512 row tiles.
//   Each wave handles column tiles ct = w, w+8, ... (64 iterations).
// ---------------------------------------------------------------------------
__global__ __launch_bounds__(256) void triphard_main(
    const float* __restrict__ inp, const float* __restrict__ pos,
    const float* __restrict__ sq, float* __restrict__ partial) {

    __shared__ float lds_dist[WAVES][16 * 17];   // padded 16x16 dist tiles
    __shared__ float mergeV[16][16][5];          // [row][slot][k]
    __shared__ int   mergeI[16][16][5];
    __shared__ int   negIdx[16];
    __shared__ float hinge[16];

    const int tid  = threadIdx.x;
    const int w    = tid >> 5;
    const int lane = tid & 31;
    const int r    = lane & 15;    // row/col within tile
    const int lh   = lane >> 4;    // lane half (K/col-half selector)
    const int rowbase = blockIdx.x * 16;

    // --- A tile, resident in VGPRs.  f32 16x16x4 WMMA layout:
    //     lane (r, lh) chunk c holds A[r][4c + 2*lh + {0,1}]
    v2f a2[32];
    {
        const float* arow = inp + (rowbase + r) * DIM + 2 * lh;
#pragma unroll
        for (int c = 0; c < 32; ++c) a2[c] = *(const v2f*)(arow + 4 * c);
    }

    float sqA[8];
#pragma unroll
    for (int v = 0; v < 8; ++v) sqA[v] = sq[rowbase + 8 * lh + v];

    float tv[5]; int ti[5];
#pragma unroll
    for (int s = 0; s < 5; ++s) { tv[s] = 3.4e38f; ti[s] = 0x7fffffff; }

    for (int ct = w; ct < NTILE; ct += WAVES) {
        const int colbase = ct * 16;

        // prefetch this wave's next column tile into the cache hierarchy
        const int ctn = ct + WAVES;
        if (ctn < NTILE)
            __builtin_prefetch((const void*)(inp + (ctn * 16 + r) * DIM + lh * 64), 0, 3);

        // B tile: identical per-lane pattern (Gram matrix, B = A^T)
        v2f b2[32];
        {
            const float* brow = inp + (colbase + r) * DIM + 2 * lh;
#pragma unroll
            for (int c = 0; c < 32; ++c) b2[c] = *(const v2f*)(brow + 4 * c);
        }

        // 32 chained f32 WMMAs, 4 accumulators to hide matrix-pipe latency
        v8f acc0 = {}, acc1 = {}, acc2 = {}, acc3 = {};
#pragma unroll
        for (int c = 0; c < 32; c += 4) {
            acc0 = __builtin_amdgcn_wmma_f32_16x16x4_f32(false, a2[c + 0], false, b2[c + 0], (short)0, acc0, false, false);
            acc1 = __builtin_amdgcn_wmma_f32_16x16x4_f32(false, a2[c + 1], false, b2[c + 1], (short)0, acc1, false, false);
            acc2 = __builtin_amdgcn_wmma_f32_16x16x4_f32(false, a2[c + 2], false, b2[c + 2], (short)0, acc2, false, false);
            acc3 = __builtin_amdgcn_wmma_f32_16x16x4_f32(false, a2[c + 3], false, b2[c + 3], (short)0, acc3, false, false);
        }
        v8f acc = (acc0 + acc1) + (acc2 + acc3);

        // dist^2 = sq[m] + sq[n] - 2*G ; transpose through LDS (C layout is
        // column-striped across lanes, selection needs row-per-lane).
        const float sqn = sq[colbase + r];
#pragma unroll
        for (int v = 0; v < 8; ++v) {
            float d2 = fmaxf(sqA[v] + sqn - 2.0f * acc[v], 1e-12f);
            lds_dist[w][(8 * lh + v) * 17 + r] = d2;
        }
        __builtin_amdgcn_wave_barrier();
        asm volatile("s_wait_dscnt 0x0" ::: "memory");

        // lane owns row r, column half lh (8 candidates per iteration)
        const int j0 = lh * 8;
#pragma unroll
        for (int j = 0; j < 8; ++j) {
            float d = lds_dist[w][r * 17 + j0 + j];
            top5_insert(tv, ti, d, colbase + j0 + j);
        }
        __builtin_amdgcn_wave_barrier();
    }

    // --- merge 16 partial top-5 lists per row (8 waves x 2 half-lanes) ---
    const int slot = w * 2 + lh;
#pragma unroll
    for (int s = 0; s < 5; ++s) {
        mergeV[r][slot][s] = tv[s];
        mergeI[r][slot][s] = ti[s];
    }
    __syncthreads();

    if (tid < 16) {
        float fv[5]; int fi[5];
#pragma unroll
        for (int s = 0; s < 5; ++s) { fv[s] = 3.4e38f; fi[s] = 0x7fffffff; }
        for (int sl = 0; sl < 16; ++sl)
#pragma unroll
            for (int s = 0; s < 5; ++s)
                top5_insert(fv, fi, mergeV[tid][sl][s], mergeI[tid][sl][s]);
        negIdx[tid] = fi[4];   // 5th-nearest (includes self at ~0 distance)
    }
    __syncthreads();

    // --- triplet hinge: wave w handles rows 2w, 2w+1; lane l -> 4 elems ---
#pragma unroll
    for (int rr = 0; rr < 2; ++rr) {
        const int row = w * 2 + rr;
        const int g = rowbase + row;
        const float* xi = inp + (size_t)g * DIM;
        const float* xp = pos + (size_t)g * DIM;
        const float* xn = inp + (size_t)negIdx[row] * DIM;
        const float4 vi = *(const float4*)(xi + lane * 4);
        const float4 vp = *(const float4*)(xp + lane * 4);
        const float4 vn = *(const float4*)(xn + lane * 4);
        const float eps = 1e-6f;
        float dp, dn, sp = 0.0f, sn = 0.0f;
        dp = vi.x - vp.x + eps; sp += dp * dp; dn = vi.x - vn.x + eps; sn += dn * dn;
        dp = vi.y - vp.y + eps; sp += dp * dp; dn = vi.y - vn.y + eps; sn += dn * dn;
        dp = vi.z - vp.z + eps; sp += dp * dp; dn = vi.z - vn.z + eps; sn += dn * dn;
        dp = vi.w - vp.w + eps; sp += dp * dp; dn = vi.w - vn.w + eps; sn += dn * dn;
#pragma unroll
        for (int m = 16; m >= 1; m >>= 1) {
            sp += __shfl_xor(sp, m, 32);
            sn += __shfl_xor(sn, m, 32);
        }
        if (lane == 0)
            hinge[row] = fmaxf(sqrtf(sp) - sqrtf(sn) + 0.3f, 0.0f);
    }
    __syncthreads();

    if (tid == 0) {
        float s = 0.0f;
        for (int i = 0; i < 16; ++i) s += hinge[i];
        partial[blockIdx.x] = s;
    }
}

// ---------------------------------------------------------------------------
// Kernel 3: deterministic final reduction of 512 block partials -> mean
// ---------------------------------------------------------------------------
__global__ __launch_bounds__(256) void triphard_reduce(
    const float* __restrict__ partial, float* __restrict__ out) {
    __shared__ float buf[256];
    const int t = threadIdx.x;
    buf[t] = partial[t] + partial[t + 256];
    __syncthreads();
    for (int off = 128; off > 0; off >>= 1) {
        if (t < off) buf[t] += buf[t + off];
        __syncthreads();
    }
    if (t == 0) out[0] = buf[0] * (1.0f / 8192.0f);
}

// ---------------------------------------------------------------------------
extern "C" void kernel_launch(void* const* d_in, const int* in_sizes, int n_in,
                              void* d_out, int out_size, void* d_ws, size_t ws_size,
                              hipStream_t stream) {
    const float* inputs   = (const float*)d_in[0];
    const float* positive = (const float*)d_in[1];
    float* out = (float*)d_out;
    float* ws  = (float*)d_ws;

    float* sq      = ws;          // 8192 floats
    float* partial = ws + NPTS;   // 512 floats

    triphard_sqnorm<<<NPTS / 8, 256, 0, stream>>>(inputs, sq);
    triphard_main<<<NTILE, 256, 0, stream>>>(inputs, positive, sq, partial);
    triphard_reduce<<<1, 256, 0, stream>>>(partial, out);
}